// QDDPM_cpu_3779571220827
// MI455X (gfx1250) — compile-verified
//
#include <hip/hip_runtime.h>
#include <hip/hip_bf16.h>

// ---------------------------------------------------------------------------
// QDDPM back-circuit + ancilla sampling for MI455X (gfx1250, wave32).
//
//  * One wave32 per sample; state (4096 complex f32 = 32KB) lives in LDS
//    (4 samples / 128-thread block = 128KB < 320KB WGP LDS).
//  * Per layer, 4-qubit groups are fused into 16x16 complex unitaries and
//    applied as 16x16 x 16x256 complex matmuls on V_WMMA_F32_16X16X4_F32
//    (full fp32 precision). Complex matmul = 4 real WMMA chains; -Im(U) is
//    pre-negated in registers (f32 WMMA only supports C-negate).
//  * Two column-tiles per inner iteration -> 4 independent accumulator
//    chains to cover XDL latency.
//  * Global->LDS staging via async-to-LDS (ASYNCcnt) when available,
//    else coalesced b128 loads.
//  * Write-back split per result array so consecutive stores differ by the
//    row stride only -> merges into ds_store_2addr_stride64/b128 forms.
// ---------------------------------------------------------------------------

typedef float v2f __attribute__((ext_vector_type(2)));
typedef float v8f __attribute__((ext_vector_type(8)));
typedef int   qd_v4i __attribute__((ext_vector_type(4)));

#if __has_builtin(__builtin_amdgcn_global_load_async_to_lds_b128) && \
    __has_builtin(__builtin_amdgcn_s_wait_asynccnt)
#define QD_ASYNC_LDS 1
typedef __attribute__((address_space(1))) qd_v4i* gas_i4p;   // global int4*
typedef __attribute__((address_space(3))) qd_v4i* lds_i4p;   // LDS int4*
#endif

constexpr int NTOTC   = 12;
constexpr int DIMC    = 4096;   // 2^12
constexpr int LAYERSC = 4;
constexpr int NSAMP   = 1024;
constexpr int NOUT    = 256;    // 2^8 post-collapse amplitudes
constexpr int WPB     = 4;      // waves (samples) per block
constexpr int WAVE_FLOATS = 2 * DIMC;   // re + im per sample

__global__ __launch_bounds__(32 * WPB)
void qddpm_wmma_kernel(const float* __restrict__ in_re,
                       const float* __restrict__ in_im,
                       const float* __restrict__ params,
                       const float* __restrict__ uu,
                       float* __restrict__ out)
{
    extern __shared__ float smem[];            // WPB * (2*DIMC) floats
    const int lane = threadIdx.x & 31;
    const int wave = threadIdx.x >> 5;
    const int samp = blockIdx.x * WPB + wave;

    float* sre = smem + wave * WAVE_FLOATS;
    float* sim = sre + DIMC;

    // ---- stage state into LDS (only global read of the hot path) ----
    const float* gre = in_re + (size_t)samp * DIMC;
    const float* gim = in_im + (size_t)samp * DIMC;
#ifdef QD_ASYNC_LDS
    for (int e4 = lane; e4 < DIMC / 4; e4 += 32) {
        __builtin_amdgcn_global_load_async_to_lds_b128(
            (gas_i4p)(gre + 4 * e4), (lds_i4p)(sre + 4 * e4), 0, 0);
        __builtin_amdgcn_global_load_async_to_lds_b128(
            (gas_i4p)(gim + 4 * e4), (lds_i4p)(sim + 4 * e4), 0, 0);
    }
    __builtin_amdgcn_s_wait_asynccnt(0);
#else
    for (int e4 = lane; e4 < DIMC / 4; e4 += 32) {
        ((float4*)sre)[e4] = ((const float4*)gre)[e4];
        ((float4*)sim)[e4] = ((const float4*)gim)[e4];
    }
#endif

    const int mrow  = lane & 15;        // M / N position of this lane
    const int khalf = (lane >> 4) * 2;  // K sub-position (A/B frag layout)

    for (int lay = 0; lay < LAYERSC; ++lay) {
        const int off = 2 * NTOTC * lay;
#pragma unroll
        for (int g = 0; g < 3; ++g) {
            // ---- per-qubit 2x2 unitaries u = Ry * Rx ----
            float ur[4][2][2], ui[4][2][2];
#pragma unroll
            for (int k = 0; k < 4; ++k) {
                const int q = 4 * g + k;
                const float thx = params[off + q] * 0.5f;
                const float thy = params[off + NTOTC + q] * 0.5f;
                const float cx = __cosf(thx), sx = __sinf(thx);
                const float cy = __cosf(thy), sy = __sinf(thy);
                ur[k][0][0] =  cy * cx;  ui[k][0][0] =  sy * sx;
                ur[k][0][1] = -sy * cx;  ui[k][0][1] = -cy * sx;
                ur[k][1][0] =  sy * cx;  ui[k][1][0] = -cy * sx;
                ur[k][1][1] =  cy * cx;  ui[k][1][1] = -sy * sx;
            }

            // ---- A fragments of the 16x16 group unitary (analytic) ----
            // U[i][j] = prod_k u_k[bit(i,3-k)][bit(j,3-k)]
            v2f aRe[4], aIm[4], aImN[4];
#pragma unroll
            for (int kc = 0; kc < 4; ++kc) {
#pragma unroll
                for (int t = 0; t < 2; ++t) {
                    const int kk = kc * 4 + khalf + t;
                    float pr = 1.0f, pi = 0.0f;
#pragma unroll
                    for (int k = 0; k < 4; ++k) {
                        const int ib = (mrow >> (3 - k)) & 1;
                        const int jb = (kk   >> (3 - k)) & 1;
                        const float wr = ur[k][ib][jb];
                        const float wi = ui[k][ib][jb];
                        const float nr = pr * wr - pi * wi;
                        pi = pr * wi + pi * wr;
                        pr = nr;
                    }
                    aRe[kc][t]  = pr;
                    aIm[kc][t]  = pi;
                    aImN[kc][t] = -pi;   // pre-negate: f32 WMMA has no A-negate
                }
            }

            // state as [l,16,r]: r = 2^(8-4g); element (j,n) at
            // ((n&~rmask)<<4) | (j<<s) | (n&rmask), n in [0,256)
            const int s     = 8 - 4 * g;
            const int rmask = (1 << s) - 1;

            // two column-tiles per iteration -> 4 independent WMMA chains
            for (int tile = 0; tile < 8; ++tile) {
                const int n0  = tile * 16 + mrow;
                const int n1  = n0 + 128;
                const int cb0 = ((n0 & ~rmask) << 4) | (n0 & rmask);
                const int cb1 = ((n1 & ~rmask) << 4) | (n1 & rmask);

                v2f bRe0[4], bIm0[4], bRe1[4], bIm1[4];
#pragma unroll
                for (int kc = 0; kc < 4; ++kc) {
                    const int j0 = kc * 4 + khalf;
                    const int r0 = (j0 << s), r1 = ((j0 + 1) << s);
                    bRe0[kc].x = sre[cb0 + r0];  bRe0[kc].y = sre[cb0 + r1];
                    bIm0[kc].x = sim[cb0 + r0];  bIm0[kc].y = sim[cb0 + r1];
                    bRe1[kc].x = sre[cb1 + r0];  bRe1[kc].y = sre[cb1 + r1];
                    bIm1[kc].x = sim[cb1 + r0];  bIm1[kc].y = sim[cb1 + r1];
                }

                v8f cRe0 = {}, cIm0 = {}, cRe1 = {}, cIm1 = {};
#pragma unroll
                for (int kc = 0; kc < 4; ++kc) {
                    cRe0 = __builtin_amdgcn_wmma_f32_16x16x4_f32(
                        false, aRe[kc],  false, bRe0[kc], (short)0, cRe0, false, false);
                    cRe1 = __builtin_amdgcn_wmma_f32_16x16x4_f32(
                        false, aRe[kc],  false, bRe1[kc], (short)0, cRe1, false, false);
                    cIm0 = __builtin_amdgcn_wmma_f32_16x16x4_f32(
                        false, aRe[kc],  false, bIm0[kc], (short)0, cIm0, false, false);
                    cIm1 = __builtin_amdgcn_wmma_f32_16x16x4_f32(
                        false, aRe[kc],  false, bIm1[kc], (short)0, cIm1, false, false);
                    cRe0 = __builtin_amdgcn_wmma_f32_16x16x4_f32(
                        false, aImN[kc], false, bIm0[kc], (short)0, cRe0, false, false);
                    cRe1 = __builtin_amdgcn_wmma_f32_16x16x4_f32(
                        false, aImN[kc], false, bIm1[kc], (short)0, cRe1, false, false);
                    cIm0 = __builtin_amdgcn_wmma_f32_16x16x4_f32(
                        false, aIm[kc],  false, bRe0[kc], (short)0, cIm0, false, false);
                    cIm1 = __builtin_amdgcn_wmma_f32_16x16x4_f32(
                        false, aIm[kc],  false, bRe1[kc], (short)0, cIm1, false, false);
                }

                // D store: VGPR v -> row (v + 8*(lane>=16)), col = lane&15.
                // One loop per result array: consecutive stores differ only by
                // the row stride (1<<s)*4B -> mergeable ds_store_2addr/b128.
                const int mbase = (lane >> 4) * 8;
#pragma unroll
                for (int v = 0; v < 8; ++v)
                    sre[cb0 + ((mbase + v) << s)] = cRe0[v];
#pragma unroll
                for (int v = 0; v < 8; ++v)
                    sim[cb0 + ((mbase + v) << s)] = cIm0[v];
#pragma unroll
                for (int v = 0; v < 8; ++v)
                    sre[cb1 + ((mbase + v) << s)] = cRe1[v];
#pragma unroll
                for (int v = 0; v < 8; ++v)
                    sim[cb1 + ((mbase + v) << s)] = cIm1[v];
            }
        }

        // ---- CZ brick diagonal: parity of adjacent-bit AND pairs ----
#pragma unroll 4
        for (int e = lane; e < DIMC; e += 32) {
            const float sg = (__popc(e & (e >> 1)) & 1) ? -1.0f : 1.0f;
            sre[e] *= sg;
            sim[e] *= sg;
        }
    }

    // ---- ancilla outcome probabilities (16 outcomes x 256 amplitudes) ----
    float p[16];
#pragma unroll
    for (int m = 0; m < 16; ++m) {
        float acc = 0.0f;
#pragma unroll 2
        for (int t = lane; t < NOUT; t += 32) {
            const float re = sre[m * NOUT + t];
            const float im = sim[m * NOUT + t];
            acc += re * re + im * im;
        }
#pragma unroll
        for (int o = 16; o > 0; o >>= 1)
            acc += __shfl_xor(acc, o, 32);
        p[m] = acc;                       // every lane holds the full sum
    }

    float total = 0.0f;
#pragma unroll
    for (int m = 0; m < 16; ++m) total += p[m];

    // ---- inverse-CDF multinomial sample (uniform across the wave) ----
    const float thresh = uu[samp] * total;
    int   msel  = 15;
    float selp  = 0.0f;
    float cum   = 0.0f;
    bool  found = false;
#pragma unroll
    for (int m = 0; m < 16; ++m) {
        cum += p[m];
        if (!found && (cum >= thresh || m == 15)) {
            msel  = m;
            selp  = p[m];
            found = true;
        }
    }
    const float inv_norm = rsqrtf(selp);

    // ---- collapsed, normalized state -> out[samp][k][{re,im}] ----
    float2* o = (float2*)(out + (size_t)samp * (2 * NOUT));
#pragma unroll 2
    for (int t = lane; t < NOUT; t += 32) {
        float2 v;
        v.x = sre[msel * NOUT + t] * inv_norm;
        v.y = sim[msel * NOUT + t] * inv_norm;
        o[t] = v;
    }
}

extern "C" void kernel_launch(void* const* d_in, const int* in_sizes, int n_in,
                              void* d_out, int out_size, void* d_ws, size_t ws_size,
                              hipStream_t stream) {
    (void)in_sizes; (void)n_in; (void)out_size; (void)d_ws; (void)ws_size;
    const float* in_re  = (const float*)d_in[0];
    const float* in_im  = (const float*)d_in[1];
    const float* params = (const float*)d_in[2];
    const float* u      = (const float*)d_in[3];
    float* out = (float*)d_out;

    const dim3 grid(NSAMP / WPB);                 // 256 workgroups
    const dim3 block(32 * WPB);                   // 128 threads = 4 wave32
    const size_t smem_bytes = (size_t)WPB * WAVE_FLOATS * sizeof(float); // 128 KB

    qddpm_wmma_kernel<<<grid, block, smem_bytes, stream>>>(in_re, in_im, params, u, out);
}